// VFE_66649302499768
// MI455X (gfx1250) — compile-verified
//
#include <hip/hip_runtime.h>
#include <stdint.h>

typedef __attribute__((ext_vector_type(2))) float v2f;
typedef __attribute__((ext_vector_type(8))) float v8f;

#define D_IN   10
#define D_OUT  64
#define KPAD   12
#define NPAIR  55          // upper triangle of 10x10
#define NSTAT  (D_IN + NPAIR)
#define WS_WSMAT 80                    // float offset of folded W' [12][64]
#define WS_BVEC  (WS_WSMAT + KPAD*D_OUT)

// order-preserving float -> u32 key, branchless (key==0 unreachable for finite f)
__device__ __forceinline__ unsigned f2key(float f) {
    unsigned b = __float_as_uint(f);
    return b ^ (unsigned)(((int)b >> 31) | 0x80000000);
}

// ---------------- Kernel 0: zero output keys + stats accumulators ----------
__global__ void k0_zero(uint4* __restrict__ out4, int n4, float* __restrict__ ws) {
    int i = blockIdx.x * blockDim.x + threadIdx.x;
    int stride = gridDim.x * blockDim.x;
    uint4 z = {0u, 0u, 0u, 0u};
    for (int j = i; j < n4; j += stride) out4[j] = z;
    if (i < WS_WSMAT) ws[i] = 0.0f;
}

// ---------------- Kernel 1: S1[10] and upper-tri S2[55] of x --------------
__global__ void __launch_bounds__(256)
k1_stats(const float* __restrict__ x, int N, float* __restrict__ stats) {
    float s1[D_IN];
    float s2[NPAIR];
#pragma unroll
    for (int p = 0; p < D_IN; p++) s1[p] = 0.f;
#pragma unroll
    for (int p = 0; p < NPAIR; p++) s2[p] = 0.f;

    int stride = gridDim.x * blockDim.x;
    for (int row = blockIdx.x * blockDim.x + threadIdx.x; row < N; row += stride) {
        const float* xr = x + (size_t)row * D_IN;
        float xv[D_IN];
#pragma unroll
        for (int p = 0; p < D_IN; p += 2) {      // 8B-aligned: 40*row + 4*p, p even
            v2f t = *(const v2f*)(xr + p);
            xv[p] = t.x; xv[p + 1] = t.y;
        }
#pragma unroll
        for (int p = 0; p < D_IN; p++) s1[p] += xv[p];
        int c = 0;
#pragma unroll
        for (int p = 0; p < D_IN; p++)
#pragma unroll
            for (int q = p; q < D_IN; q++) s2[c++] += xv[p] * xv[q];
    }

    // wave32 reduction
#pragma unroll
    for (int p = 0; p < D_IN; p++)
        for (int off = 16; off; off >>= 1) s1[p] += __shfl_down(s1[p], off, 32);
#pragma unroll
    for (int p = 0; p < NPAIR; p++)
        for (int off = 16; off; off >>= 1) s2[p] += __shfl_down(s2[p], off, 32);

    __shared__ float red[NSTAT];
    if (threadIdx.x < NSTAT) red[threadIdx.x] = 0.f;
    __syncthreads();
    if ((threadIdx.x & 31) == 0) {
#pragma unroll
        for (int p = 0; p < D_IN; p++) atomicAdd(&red[p], s1[p]);
#pragma unroll
        for (int p = 0; p < NPAIR; p++) atomicAdd(&red[D_IN + p], s2[p]);
    }
    __syncthreads();
    if (threadIdx.x < NSTAT) atomicAdd(&stats[threadIdx.x], red[threadIdx.x]);
}

// ------- Kernel 2: fold BN into W' = W*diag(a), b' = beta - mean*a ---------
__global__ void k2_fold(const float* __restrict__ W, const float* __restrict__ gamma,
                        const float* __restrict__ beta, const float* __restrict__ stats,
                        float* __restrict__ Ws, float* __restrict__ bvec, float invN) {
    int j = threadIdx.x;
    if (j >= D_OUT) return;
    float wj[D_IN];
#pragma unroll
    for (int p = 0; p < D_IN; p++) wj[p] = W[p * D_OUT + j];

    float mean = 0.f;
#pragma unroll
    for (int p = 0; p < D_IN; p++) mean += stats[p] * invN * wj[p];

    float ex2 = 0.f;
    int c = 0;
#pragma unroll
    for (int p = 0; p < D_IN; p++)
#pragma unroll
        for (int q = p; q < D_IN; q++) {
            float term = stats[D_IN + c] * invN * wj[p] * wj[q];
            ex2 += (p == q) ? term : 2.f * term;
            c++;
        }
    float var = ex2 - mean * mean;
    var = var < 0.f ? 0.f : var;
    float a = gamma[j] * rsqrtf(var + 1e-3f);
#pragma unroll
    for (int p = 0; p < D_IN; p++) Ws[p * D_OUT + j] = wj[p] * a;
    Ws[10 * D_OUT + j] = 0.f;     // K padding rows
    Ws[11 * D_OUT + j] = 0.f;
    bvec[j] = beta[j] - mean * a;
}

// ---- Kernel 3: WMMA f32 GEMM (16 rows x 64 cols per wave) + scatter-max ---
// Full 16-row tiles only: no per-element row guards anywhere.
__global__ void __launch_bounds__(256)
k3_gemm_scatter(const float* __restrict__ x, const int* __restrict__ seg,
                const float* __restrict__ Ws, const float* __restrict__ bvec,
                unsigned* __restrict__ out_u, int nFull) {
    const int lane = threadIdx.x & 31;
    const int hi = lane >> 4;     // 0/1
    const int lo = lane & 15;

    // Preload B fragments (4 col-tiles x 3 K-steps), reused for all row tiles.
    // B 4x16 f32 layout: VGPR0 = {K=k0 (lanes 0-15), K=k0+2 (lanes 16-31)},
    //                    VGPR1 = {K=k0+1, K=k0+3}; N = lane&15.
    v2f bfrag[4][3];
#pragma unroll
    for (int t = 0; t < 4; t++)
#pragma unroll
        for (int s = 0; s < 3; s++) {
            int krow = 4 * s + hi * 2;
            int col = 16 * t + lo;
            v2f b;
            b.x = Ws[krow * D_OUT + col];
            b.y = Ws[(krow + 1) * D_OUT + col];
            bfrag[t][s] = b;
        }
    float bb[4];
#pragma unroll
    for (int t = 0; t < 4; t++) bb[t] = bvec[16 * t + lo];

    int waveId = (blockIdx.x * blockDim.x + threadIdx.x) >> 5;
    int nWaves = (gridDim.x * blockDim.x) >> 5;
    const int kofs = hi * 2;

    for (int tile = waveId; tile < nFull; tile += nWaves) {
        int rowBase = tile << 4;

        // A 16x4 f32 layout: lane<16 holds {K=k0,k0+1} of row M=lane,
        //                    lane>=16 holds {K=k0+2,k0+3} of row M=lane-16.
        const float* xr = x + (size_t)(rowBase + lo) * D_IN;
        v2f afrag[3];
        afrag[0] = *(const v2f*)(xr + kofs);          // k = 0/2 , 1/3
        afrag[1] = *(const v2f*)(xr + 4 + kofs);      // k = 4/6 , 5/7
        v2f a2 = *(const v2f*)(xr + 8);               // k = 8,9 (in-bounds for all)
        if (hi) { a2.x = 0.f; a2.y = 0.f; }           // k = 10,11 pad -> cndmask
        afrag[2] = a2;

        // seg ids for my 8 rows: contiguous 32B, broadcast across the half-wave
        const int* sp = seg + rowBase + hi * 8;       // 16B aligned
        int4 sA = *(const int4*)sp;
        int4 sB = *(const int4*)(sp + 4);
        int segBase[8] = { sA.x << 6, sA.y << 6, sA.z << 6, sA.w << 6,
                           sB.x << 6, sB.y << 6, sB.z << 6, sB.w << 6 };

#pragma unroll
        for (int t = 0; t < 4; t++) {
            v8f acc = {};
#pragma unroll
            for (int s = 0; s < 3; s++)
                acc = __builtin_amdgcn_wmma_f32_16x16x4_f32(
                    false, afrag[s], false, bfrag[t][s], (short)0, acc, false, false);

            int col = 16 * t + lo;

            // Phase 1: all keys + offsets (pure VALU, no memory)
            unsigned key[8];
            int off[8];
#pragma unroll
            for (int r = 0; r < 8; r++) {
                key[r] = f2key(acc[r] + bb[t]);
                off[r] = segBase[r] + col;            // 32-bit: saddr+voffset form
            }

            // Phase 2: batch all 8 filter loads -> 8-deep MLP, one wait total
            unsigned cur[8];
#pragma unroll
            for (int r = 0; r < 8; r++)
                cur[r] = out_u[off[r]];

            // Phase 3: conditional no-return atomics (~15% taken)
#pragma unroll
            for (int r = 0; r < 8; r++)
                if (key[r] > cur[r]) atomicMax(&out_u[off[r]], key[r]);
        }
    }
}

// ---- Kernel 3b: scalar tail for the <16 remainder rows --------------------
__global__ void k3_tail(const float* __restrict__ x, const int* __restrict__ seg,
                        const float* __restrict__ Ws, const float* __restrict__ bvec,
                        unsigned* __restrict__ out_u, int rowStart, int N) {
    int idx = blockIdx.x * blockDim.x + threadIdx.x;
    int total = (N - rowStart) * D_OUT;
    if (idx >= total) return;
    int row = rowStart + idx / D_OUT;
    int col = idx % D_OUT;
    float acc = bvec[col];
#pragma unroll
    for (int k = 0; k < D_IN; k++)
        acc += x[(size_t)row * D_IN + k] * Ws[k * D_OUT + col];
    unsigned key = f2key(acc);
    int off = (seg[row] << 6) + col;
    if (key > out_u[off]) atomicMax(&out_u[off], key);
}

// ---------------- Kernel 4: decode keys (0 marker == empty -> 0.0) ---------
__global__ void k4_decode(unsigned* __restrict__ out_u, float* __restrict__ out_f, int n) {
    int i = blockIdx.x * blockDim.x + threadIdx.x;
    int stride = gridDim.x * blockDim.x;
    for (int j = i; j < n; j += stride) {
        unsigned k = out_u[j];
        float f;
        if (k == 0u) f = 0.f;
        else {
            unsigned b = (k & 0x80000000u) ? (k ^ 0x80000000u) : ~k;
            f = __uint_as_float(b);
        }
        out_f[j] = f;
    }
}

extern "C" void kernel_launch(void* const* d_in, const int* in_sizes, int n_in,
                              void* d_out, int out_size, void* d_ws, size_t ws_size,
                              hipStream_t stream) {
    const float* x     = (const float*)d_in[0];
    const float* W     = (const float*)d_in[1];
    const float* gamma = (const float*)d_in[2];
    const float* beta  = (const float*)d_in[3];
    const int*   seg   = (const int*)d_in[4];
    int N = in_sizes[4];                        // 4,000,000 rows

    float* ws    = (float*)d_ws;
    float* stats = ws;                           // 65 floats
    float* Ws    = ws + WS_WSMAT;                // 12*64 folded weights
    float* bvec  = ws + WS_BVEC;                 // 64 biases
    unsigned* out_u = (unsigned*)d_out;
    float*    out_f = (float*)d_out;

    int nFull = N >> 4;                          // full 16-row tiles
    int remStart = nFull << 4;

    k0_zero        <<<2048, 256, 0, stream>>>((uint4*)d_out, out_size >> 2, ws);
    k1_stats       <<<1024, 256, 0, stream>>>(x, N, stats);
    k2_fold        <<<1,    64,  0, stream>>>(W, gamma, beta, stats, Ws, bvec, 1.0f / (float)N);
    k3_gemm_scatter<<<2048, 256, 0, stream>>>(x, seg, Ws, bvec, out_u, nFull);
    if (remStart < N) {
        int tail = (N - remStart) * D_OUT;
        k3_tail    <<<(tail + 255) / 256, 256, 0, stream>>>(x, seg, Ws, bvec, out_u, remStart, N);
    }
    k4_decode      <<<4096, 256, 0, stream>>>(out_u, out_f, out_size);
}